// MMLM_66133906424193
// MI455X (gfx1250) — compile-verified
//
#include <hip/hip_runtime.h>
#include <cstdint>

typedef __attribute__((ext_vector_type(4))) float v4f;
typedef __attribute__((ext_vector_type(4))) int   v4i;

// Problem constants (match reference)
#define Bc 4
#define Qc 8
#define Tc 4096
#define Dc 1024
#define TCHUNK 8   // t-values per block (pipeline depth)

// out[b,t,d] = sum_q softmax(w)[q] * E[ids[b,q,t], d]
// Block: 256 threads = 8 independent waves (no block barrier needed).
// Wave w covers d in [w*128, w*128+128); lane owns a float4.
// 8 codebook rows per t staged via async global->LDS (saddr form: scalar row
// base + per-lane 32-bit offset), double buffered, pipelined over 8 t's.
__global__ __launch_bounds__(256) void mmlm_embed_gather_kernel(
    const int*   __restrict__ input_ids,   // [B, Q, T]
    const float* __restrict__ embed,       // [V, D]
    const float* __restrict__ audio_w,     // [Q, 1]
    float*       __restrict__ out)         // [B, T, D]
{
    // staging: [buf][wave][q][lane] v4f -> 2*8*8*32*16 = 64 KB
    __shared__ v4f stage[2][8][Qc][32];

    const int tid = threadIdx.x;
    const int w   = tid >> 5;     // wave id in block
    const int l   = tid & 31;     // lane id

    const int blocksPerB = Tc / TCHUNK;          // 512
    const int b     = blockIdx.x / blocksPerB;
    const int tbase = (blockIdx.x % blocksPerB) * TCHUNK;

    // Token ids for this block: wave-uniform. Two vector loads per codebook,
    // values forced scalar at use via readfirstlane.
    int idbuf[TCHUNK][Qc];
    #pragma unroll
    for (int q = 0; q < Qc; ++q) {
        const v4i* p4 = (const v4i*)(input_ids + ((size_t)(b * Qc + q)) * Tc + tbase);
        v4i a = p4[0];
        v4i c = p4[1];
        idbuf[0][q] = a.x; idbuf[1][q] = a.y; idbuf[2][q] = a.z; idbuf[3][q] = a.w;
        idbuf[4][q] = c.x; idbuf[5][q] = c.y; idbuf[6][q] = c.z; idbuf[7][q] = c.w;
    }

    // softmax over the 8 codebook weights (trivial)
    float p[Qc];
    {
        float m = -3.402823466e38f;
        #pragma unroll
        for (int q = 0; q < Qc; ++q) { p[q] = audio_w[q]; m = fmaxf(m, p[q]); }
        float sum = 0.f;
        #pragma unroll
        for (int q = 0; q < Qc; ++q) { p[q] = __expf(p[q] - m); sum += p[q]; }
        const float inv = 1.f / sum;
        #pragma unroll
        for (int q = 0; q < Qc; ++q) p[q] *= inv;
    }

    // Per-lane invariants.
    const unsigned voff = (unsigned)(tid * 16);                     // byte offset into a row
    const unsigned lds0 = (unsigned)(uintptr_t)&stage[0][w][0][l];  // low32 = LDS offset
    const uint64_t ebase = (uint64_t)(uintptr_t)embed;
    const int d_off = w * 128 + l * 4;

    // Issue one pipeline stage: 8 async b128 gathers (scalar base per row).
    auto issue = [&](int s, int buf) {
        #pragma unroll
        for (int q = 0; q < Qc; ++q) {
            const unsigned row = (unsigned)__builtin_amdgcn_readfirstlane(idbuf[s][q]);
            const uint64_t base = ebase + ((uint64_t)row << 12);   // row * D * 4
            const unsigned lds_addr = lds0 + (unsigned)(buf * 32768 + q * 512);
            asm volatile("global_load_async_to_lds_b128 %0, %1, %2"
                         :: "v"(lds_addr), "v"(voff), "s"(base) : "memory");
        }
    };

    issue(0, 0);   // prologue

    #pragma unroll
    for (int s = 0; s < TCHUNK; ++s) {
        const int buf = s & 1;
        if (s + 1 < TCHUNK) {
            issue(s + 1, (s + 1) & 1);
            // async loads complete in order: <=8 outstanding => stage s landed
            asm volatile("s_wait_asynccnt 0x8" ::: "memory");
        } else {
            asm volatile("s_wait_asynccnt 0x0" ::: "memory");
        }

        v4f acc = (v4f)(0.f);
        #pragma unroll
        for (int q = 0; q < Qc; ++q) {
            v4f v = stage[buf][w][q][l];         // ds_load_b128
            acc.x = fmaf(p[q], v.x, acc.x);
            acc.y = fmaf(p[q], v.y, acc.y);
            acc.z = fmaf(p[q], v.z, acc.z);
            acc.w = fmaf(p[q], v.w, acc.w);
        }

        const int t = tbase + s;
        v4f* dst = (v4f*)(out + ((size_t)b * Tc + t) * Dc + d_off);
        __builtin_nontemporal_store(acc, dst);   // write-once stream: keep L2 for table
    }
}

extern "C" void kernel_launch(void* const* d_in, const int* in_sizes, int n_in,
                              void* d_out, int out_size, void* d_ws, size_t ws_size,
                              hipStream_t stream)
{
    (void)in_sizes; (void)n_in; (void)out_size; (void)d_ws; (void)ws_size;
    const int*   input_ids = (const int*)  d_in[0];   // [B,Q,T] int32
    const float* embed     = (const float*)d_in[1];   // [V,D]   f32
    const float* audio_w   = (const float*)d_in[2];   // [Q,1]   f32
    float*       out       = (float*)d_out;           // [B,T,D] f32

    const int grid = Bc * (Tc / TCHUNK);              // 2048 blocks
    mmlm_embed_gather_kernel<<<grid, 256, 0, stream>>>(input_ids, embed, audio_w, out);
}